// BEVFusionModel_18133351923977
// MI455X (gfx1250) — compile-verified
//
#include <hip/hip_runtime.h>
#include <hip/hip_bf16.h>

// ---- problem constants (from reference) ----
#define NCAM   6
#define DD     48
#define HF     64
#define WF     176
#define CC     80
#define BEV_H  128
#define BEV_W  128
#define BEV_Z  8
#define BEV_C  128
#define HW     (BEV_H * BEV_W)          // 16384
#define KDIM   (CC * BEV_Z)             // 640
#define NPTS   (NCAM * DD * HF * WF)    // 3,244,032
#define HWC    (HF * WF)                // 11264

typedef __attribute__((ext_vector_type(2))) float v2f;
typedef __attribute__((ext_vector_type(8))) float v8f;

// -------------------------------------------------------------------------
// Kernel 1: project rays, outer-product depth*context, scatter-add into the
// voxel accumulator laid out as voxT[k = c*8 + z][hw] (GEMM-friendly), plus
// per-voxel hit counts cnt[z*HW + hw].
// -------------------------------------------------------------------------
__global__ __launch_bounds__(256)
void bev_scatter_kernel(const float* __restrict__ depth_prob,
                        const float* __restrict__ context,
                        const float* __restrict__ intr,
                        const float* __restrict__ c2e,
                        float* __restrict__ voxT,
                        float* __restrict__ cnt)
{
    unsigned gid = blockIdx.x * 256u + threadIdx.x;
    if (gid >= NPTS) return;

    unsigned w =  gid % WF;
    unsigned h = (gid / WF) % HF;
    unsigned d = (gid / (WF * HF)) % DD;
    unsigned n =  gid / (WF * HF * DD);

    float u  = ((float)w + 0.5f) * 4.0f;      // STRIDE = 4
    float vv = ((float)h + 0.5f) * 4.0f;
    float Z  = 1.0f + (float)d * (59.0f / 47.0f);   // linspace(1,60,48)

    const float* Kn = intr + n * 9;
    float Xc = (u  - Kn[2]) / Kn[0] * Z;
    float Yc = (vv - Kn[5]) / Kn[4] * Z;

    const float* Tn = c2e + n * 16;
    float Xe = Tn[0]*Xc + Tn[1]*Yc + Tn[2] *Z + Tn[3];
    float Ye = Tn[4]*Xc + Tn[5]*Yc + Tn[6] *Z + Tn[7];
    float Ze = Tn[8]*Xc + Tn[9]*Yc + Tn[10]*Z + Tn[11];

    // mx = my = 100/128 -> inv = 1.28 exactly; mz = 1.0
    int ix = (int)floorf((Xe + 50.0f) * 1.28f);
    int iy = (int)floorf((Ye + 50.0f) * 1.28f);
    int iz = (int)floorf( Ze + 5.0f);

    if (ix < 0 || ix >= BEV_W || iy < 0 || iy >= BEV_H ||
        iz < 0 || iz >= BEV_Z) return;

    int hw = iy * BEV_W + ix;
    atomicAdd(cnt + iz * HW + hw, 1.0f);

    float dp = depth_prob[gid];                        // (n,d,h,w) flat == gid
    const float* ctx = context + (n * CC * HF + h) * WF + w;
    float* vt = voxT + iz * HW + hw;                   // + c*(8*HW) per channel

#pragma unroll 4
    for (int c = 0; c < CC; ++c) {
        atomicAdd(vt + c * (BEV_Z * HW), dp * ctx[c * HWC]);
    }
}

// -------------------------------------------------------------------------
// Kernel 2: voxT[k][hw] /= max(cnt[z][hw], 1)  (in place)
// -------------------------------------------------------------------------
__global__ __launch_bounds__(256)
void bev_normalize_kernel(float* __restrict__ voxT,
                          const float* __restrict__ cnt)
{
    unsigned idx = blockIdx.x * 256u + threadIdx.x;   // over KDIM*HW
    if (idx >= (unsigned)(KDIM * HW)) return;
    unsigned k  = idx >> 14;          // /HW
    unsigned hw = idx & (HW - 1);
    unsigned z  = k & 7;
    float c = cnt[z * HW + hw];
    voxT[idx] /= fmaxf(c, 1.0f);
}

// -------------------------------------------------------------------------
// Kernel 3: bev = relu( BN( w_proj @ voxT ) ), fp32 WMMA 16x16x4.
//   grid = (HW/16/8, BEV_C/16); 256 threads = 8 waves; each wave owns one
//   16(out-ch) x 16(spatial) tile; all waves in a block share the A tile,
//   staged in LDS with pitch 642 to avoid 64-bank conflicts (stride 640).
// -------------------------------------------------------------------------
#define SA_PITCH 642

__global__ __launch_bounds__(256)
void bev_gemm_bn_relu_wmma(const float* __restrict__ Wp,
                           const float* __restrict__ voxT,
                           const float* __restrict__ gamma,
                           const float* __restrict__ beta,
                           const float* __restrict__ mean,
                           const float* __restrict__ var,
                           float* __restrict__ out)
{
    __shared__ float sA[16 * SA_PITCH];

    const int tid    = threadIdx.x;
    const int o_base = blockIdx.y * 16;

    // w_proj rows o_base..o_base+15 are contiguous (row-major 128x640)
    const float* wsrc = Wp + o_base * KDIM;
    for (int i = tid; i < 16 * KDIM; i += 256) {
        int r = i / KDIM, col = i - r * KDIM;
        sA[r * SA_PITCH + col] = wsrc[i];
    }
    __syncthreads();

    const int wave    = tid >> 5;
    const int lane    = tid & 31;
    const int hw_base = (blockIdx.x * 8 + wave) * 16;

    const int m     = lane & 15;            // A row == B/D column index
    const int khalf = (lane >> 4) << 1;     // 0 for lanes 0-15, 2 for 16-31

    const float* brow = voxT + hw_base + m;
    const float* arow = sA + m * SA_PITCH + khalf;

    v8f acc = {};
    for (int kk = 0; kk < KDIM; kk += 4) {
        v2f a = *(const v2f*)(arow + kk);             // ds_load_b64
        v2f b;
        b.x = brow[(kk + khalf)     * HW];            // coalesced per half-wave
        b.y = brow[(kk + khalf + 1) * HW];
        acc = __builtin_amdgcn_wmma_f32_16x16x4_f32(
                  false, a, false, b, (short)0, acc, false, false);
    }

    // epilogue: BN + ReLU, D layout: vgpr v -> M = v + (lane>=16 ? 8 : 0)
    const int half = (lane >> 4) << 3;
#pragma unroll
    for (int v = 0; v < 8; ++v) {
        int o = o_base + v + half;
        float inv  = gamma[o] * rsqrtf(var[o] + 1e-5f);
        float bias = beta[o] - mean[o] * inv;
        float val  = acc[v] * inv + bias;
        out[o * HW + hw_base + m] = fmaxf(val, 0.0f);
    }
}

// -------------------------------------------------------------------------
extern "C" void kernel_launch(void* const* d_in, const int* in_sizes, int n_in,
                              void* d_out, int out_size, void* d_ws, size_t ws_size,
                              hipStream_t stream)
{
    const float* depth_prob = (const float*)d_in[0];
    const float* context    = (const float*)d_in[1];
    const float* intr       = (const float*)d_in[2];
    const float* c2e        = (const float*)d_in[3];
    const float* w_proj     = (const float*)d_in[4];
    const float* bn_gamma   = (const float*)d_in[5];
    const float* bn_beta    = (const float*)d_in[6];
    const float* bn_mean    = (const float*)d_in[7];
    const float* bn_var     = (const float*)d_in[8];
    float* out = (float*)d_out;

    // workspace: voxT [640][16384] f32, then cnt [8*16384] f32
    const size_t VOXT_BYTES = (size_t)KDIM * HW * sizeof(float);   // 41,943,040
    const size_t CNT_BYTES  = (size_t)BEV_Z * HW * sizeof(float);  //    524,288
    float* voxT = (float*)d_ws;
    float* cnt  = (float*)((char*)d_ws + VOXT_BYTES);

    hipMemsetAsync(d_ws, 0, VOXT_BYTES + CNT_BYTES, stream);

    bev_scatter_kernel<<<(NPTS + 255) / 256, 256, 0, stream>>>(
        depth_prob, context, intr, c2e, voxT, cnt);

    bev_normalize_kernel<<<(KDIM * HW + 255) / 256, 256, 0, stream>>>(voxT, cnt);

    dim3 grid(HW / 16 / 8, BEV_C / 16);   // (128, 8)
    bev_gemm_bn_relu_wmma<<<grid, 256, 0, stream>>>(
        w_proj, voxT, bn_gamma, bn_beta, bn_mean, bn_var, out);
}